// RPE_9010841387714
// MI455X (gfx1250) — compile-verified
//
#include <hip/hip_runtime.h>
#include <hip/hip_bf16.h>
#include <cstdint>

// ---------------------------------------------------------------------------
// RPE gather kernel for MI455X (gfx1250), memory-bound:
//   reads  xyz  [B,P,P,3] int32            (~50 MB)
//   gathers rpe_table [615,16] f32 from LDS (staged once per WG via TDM)
//   writes out  [B,H,P,P] f32              (~268 MB, nontemporal)
// HBM floor ~318MB / 23.3 TB/s ~= 14 us. No matmul -> no WMMA; the CDNA5
// feature used is the Tensor Data Mover (tensor_load_to_lds + TENSORcnt).
// ---------------------------------------------------------------------------

#define BLOCK 256
#define PIX_PER_THREAD 4

typedef unsigned int       u32;
typedef unsigned long long u64;
typedef u32   v4u __attribute__((ext_vector_type(4)));
typedef int   v4i __attribute__((ext_vector_type(4)));
typedef int   v8i __attribute__((ext_vector_type(8)));
typedef float v4f __attribute__((ext_vector_type(4)));

constexpr int H         = 16;            // NUM_HEADS
constexpr int P         = 128;           // PATCH_SIZE
constexpr int POS_BND   = 102;           // 0.8 * 128
constexpr int RPE_NUM   = 2 * POS_BND + 1;   // 205
constexpr int ROWS      = 3 * RPE_NUM;       // 615
constexpr int TAB_ELEMS = ROWS * H;          // 9840 floats = 39360 B

__global__ __launch_bounds__(BLOCK) void rpe_gather_kernel(
    const int*   __restrict__ xyz,
    const float* __restrict__ rpe,
    float*       __restrict__ out,
    int n_pix)
{
    __shared__ float tab[TAB_ELEMS];

    // ---- Stage rpe_table into LDS --------------------------------------
#if __has_builtin(__builtin_amdgcn_tensor_load_to_lds)
    if (threadIdx.x < 32) {
        // Generic pointer to a __shared__ object: low 32 bits are the LDS
        // byte offset (aperture scheme, ISA 10.2).
        u32 lds_off = (u32)(u64)(uintptr_t)&tab[0];
        u64 ga      = (u64)(uintptr_t)rpe;

        // D# group 0 (ISA 8.3): count=1 (valid), lds_addr, global_addr, type=2
        v4u g0;
        g0[0] = 1u;                                   // count=1, user mode
        g0[1] = lds_off;                              // lds_addr
        g0[2] = (u32)ga;                              // global_addr[31:0]
        g0[3] = (u32)((ga >> 32) & 0x01FFFFFFu)       // global_addr[56:32]
              | (2u << 30);                           // type = 2 ("image")

        // D# group 1 (ISA 8.4): 1-D tensor of 9840 x 4-byte elements.
        v8i g1;
        g1[0] = (int)(2u << 16);                      // data_size=2 (4B); no mask/pad/iter
        g1[1] = (int)((u32)(TAB_ELEMS & 0xFFFF) << 16); // tensor_dim0 lo16 (9840)
        g1[2] = (int)(((u32)TAB_ELEMS >> 16)          // tensor_dim0 hi16 (0)
              | (1u << 16));                          // tensor_dim1 = 1
        g1[3] = (int)((u32)(TAB_ELEMS & 0xFFFF) << 16); // dim1 hi=0, tile_dim0=9840
        g1[4] = 1;                                    // tile_dim1=1, tile_dim2=0
        g1[5] = TAB_ELEMS;                            // tensor_dim0_stride lo32
        g1[6] = 0;                                    // stride hi / dim1_stride lo
        g1[7] = 0;                                    // dim1_stride hi

        v4i z4 = (v4i)0;
#if defined(__clang_major__) && (__clang_major__ >= 23)
        v8i z8 = (v8i)0;
        __builtin_amdgcn_tensor_load_to_lds(g0, g1, z4, z4, z8, 0);
#else
        __builtin_amdgcn_tensor_load_to_lds(g0, g1, z4, z4, 0);
#endif
        __builtin_amdgcn_s_wait_tensorcnt(0);
    }
#else
    for (int k = threadIdx.x; k < TAB_ELEMS; k += BLOCK) tab[k] = rpe[k];
#endif
    __syncthreads();

    // ---- Per-thread work: 4 consecutive pixels (same b, same i) --------
    long tid  = (long)blockIdx.x * BLOCK + threadIdx.x;
    long pix0 = tid * PIX_PER_THREAD;
    if (pix0 >= (long)n_pix) return;

    // xyz for 4 pixels = 12 ints = 48 contiguous bytes, 16B aligned.
    const v4i* xp = (const v4i*)(xyz + pix0 * 3);
    v4i cc[3];
    cc[0] = __builtin_nontemporal_load(xp + 0);
    cc[1] = __builtin_nontemporal_load(xp + 1);
    cc[2] = __builtin_nontemporal_load(xp + 2);

    float acc[H][PIX_PER_THREAD];
#pragma unroll
    for (int h = 0; h < H; ++h)
#pragma unroll
        for (int p = 0; p < PIX_PER_THREAD; ++p) acc[h][p] = 0.0f;

#pragma unroll
    for (int p = 0; p < PIX_PER_THREAD; ++p) {
#pragma unroll
        for (int a = 0; a < 3; ++a) {
            const int e = 3 * p + a;
            int c = cc[e >> 2][e & 3];
            c = (c < -POS_BND) ? -POS_BND : (c > POS_BND ? POS_BND : c);
            const int row = c + POS_BND + a * RPE_NUM;   // [0, 615)
            // 16 consecutive floats (64B, aligned) -> 4x ds_load_b128
            const v4f* r = (const v4f*)&tab[row * H];
            v4f q0 = r[0], q1 = r[1], q2 = r[2], q3 = r[3];
            acc[ 0][p] += q0[0]; acc[ 1][p] += q0[1]; acc[ 2][p] += q0[2]; acc[ 3][p] += q0[3];
            acc[ 4][p] += q1[0]; acc[ 5][p] += q1[1]; acc[ 6][p] += q1[2]; acc[ 7][p] += q1[3];
            acc[ 8][p] += q2[0]; acc[ 9][p] += q2[1]; acc[10][p] += q2[2]; acc[11][p] += q2[3];
            acc[12][p] += q3[0]; acc[13][p] += q3[1]; acc[14][p] += q3[2]; acc[15][p] += q3[3];
        }
    }

    // ---- Transposed store: out[b][h][i][j0..j0+3], one b128 per h ------
    const long j0 = pix0 & (P - 1);
    const long bi = pix0 >> 7;           // b*128 + i
    const long i  = bi & (P - 1);
    const long b  = bi >> 7;
    float* ob = out + b * (long)(H * P * P) + i * (long)P + j0;
#pragma unroll
    for (int h = 0; h < H; ++h) {
        v4f v = { acc[h][0], acc[h][1], acc[h][2], acc[h][3] };
        __builtin_nontemporal_store(v, (v4f*)(ob + (long)h * (P * P)));
    }
}

extern "C" void kernel_launch(void* const* d_in, const int* in_sizes, int n_in,
                              void* d_out, int out_size, void* d_ws, size_t ws_size,
                              hipStream_t stream) {
    const int*   xyz = (const int*)d_in[0];    // [B,P,P,3] int32
    const float* rpe = (const float*)d_in[1];  // [615,16]  f32
    float*       out = (float*)d_out;          // [B,H,P,P] f32

    const int n_pix     = in_sizes[0] / 3;                       // 4,194,304
    const int n_threads = (n_pix + PIX_PER_THREAD - 1) / PIX_PER_THREAD;
    const int blocks    = (n_threads + BLOCK - 1) / BLOCK;       // 4096

    rpe_gather_kernel<<<blocks, BLOCK, 0, stream>>>(xyz, rpe, out, n_pix);
}